// TopK_30520037605537
// MI455X (gfx1250) — compile-verified
//
#include <hip/hip_runtime.h>
#include <stdint.h>

#define N_TOKENS 2048
#define D_SAE    16384
#define BLOCK    512                       // 16 waves of 32
#define NWAVES   (BLOCK / 32)              // 16
#define PER_THREAD (D_SAE / BLOCK)         // 32 elements per thread
#define VEC_PER_THREAD (PER_THREAD / 4)    // 8 x uint4 per thread
#define VEC_LOADS  (D_SAE / (BLOCK * 4))   // 8 x b128 async loads per thread

__device__ __forceinline__ uint32_t f2key(uint32_t bits) {
    // monotonic map: float order -> unsigned order
    return bits ^ (uint32_t)(((int32_t)bits >> 31) | 0x80000000);
}

__global__ __launch_bounds__(BLOCK)
void topk_gate_kernel(const float* __restrict__ x,
                      const int* __restrict__ kptr,
                      float* __restrict__ out) {
    __shared__ uint32_t s_x[D_SAE];              // raw fp32 bits of the row (64 KB)
    __shared__ uint32_t s_whist[NWAVES * 256];   // per-wave histograms (16 KB)
    __shared__ uint32_t s_hist[256];             // reduced histogram / suffix sums
    __shared__ uint32_t s_sel[2];                // [0]=prefix, [1]=k remaining
    __shared__ uint32_t s_tie;

    const int tid  = threadIdx.x;
    const int wave = tid >> 5;
    const int row  = blockIdx.x;
    const int k    = *kptr;
    const size_t rowbase = (size_t)row * D_SAE;
    const float* rowp = x + rowbase;
    float* outp = out + rowbase;

    // ---- stage row HBM -> LDS via CDNA5 async copies (ASYNCcnt path) ----
    uint32_t lds_base = (uint32_t)(uintptr_t)(void*)&s_x[0];
#pragma unroll
    for (int i = 0; i < VEC_LOADS; ++i) {
        uint32_t byte_off = (uint32_t)(i * BLOCK + tid) * 16u;   // coalesced 16B/lane
        uint32_t lds_addr = lds_base + byte_off;
        asm volatile("global_load_async_to_lds_b128 %0, %1, %2"
                     :: "v"(lds_addr), "v"(byte_off), "s"(rowp)
                     : "memory");
    }
    asm volatile("s_wait_asynccnt 0" ::: "memory");
    __syncthreads();

    // ---- trivial cases (k uniform across block) ----
    if (k <= 0) {
        float4 z = make_float4(0.f, 0.f, 0.f, 0.f);
        for (int e = 0; e < VEC_PER_THREAD; ++e)
            reinterpret_cast<float4*>(outp)[e * BLOCK + tid] = z;
        return;
    }
    if (k >= D_SAE) {
        for (int e = 0; e < VEC_PER_THREAD; ++e) {
            uint4 b = reinterpret_cast<uint4*>(s_x)[e * BLOCK + tid];
            reinterpret_cast<uint4*>(outp)[e * BLOCK + tid] = b;
        }
        return;
    }

    if (tid == 0) { s_sel[0] = 0u; s_sel[1] = (uint32_t)k; s_tie = 0u; }
    __syncthreads();

    // ---- 4-pass radix select (8 bits/pass) for the k-th largest key ----
    for (int p = 0; p < 4; ++p) {
        // snapshot selection state BEFORE anyone can rewrite it this pass
        uint32_t prefix = s_sel[0];
        uint32_t krem   = s_sel[1];
        // zero per-wave histograms: 4096 words, 8 per thread
        for (int z = 0; z < (NWAVES * 256) / BLOCK; ++z)
            s_whist[z * BLOCK + tid] = 0u;
        __syncthreads();

        const int shift_hi = 32 - 8 * p;   // only used when p > 0
        const int shift_d  = 24 - 8 * p;
        uint32_t* myhist = &s_whist[wave * 256];
        for (int e = 0; e < VEC_PER_THREAD; ++e) {
            uint4 b = reinterpret_cast<uint4*>(s_x)[e * BLOCK + tid];
            uint32_t kk[4] = { f2key(b.x), f2key(b.y), f2key(b.z), f2key(b.w) };
#pragma unroll
            for (int j = 0; j < 4; ++j) {
                bool active = (p == 0) || ((kk[j] >> shift_hi) == prefix);
                if (active) atomicAdd(&myhist[(kk[j] >> shift_d) & 0xFFu], 1u);
            }
        }
        __syncthreads();

        // reduce per-wave histograms into s_hist
        if (tid < 256) {
            uint32_t s = 0u;
#pragma unroll
            for (int w = 0; w < NWAVES; ++w) s += s_whist[w * 256 + tid];
            s_hist[tid] = s;
        }
        __syncthreads();

        // parallel suffix-sum over 256 bins: s_hist[i] = count(digit >= i)
        for (int s = 1; s < 256; s <<= 1) {
            uint32_t v = 0u;
            if (tid < 256) {
                v = s_hist[tid];
                if (tid + s < 256) v += s_hist[tid + s];
            }
            __syncthreads();
            if (tid < 256) s_hist[tid] = v;
            __syncthreads();
        }

        // exactly one bin satisfies: suffix[d] >= krem > suffix[d+1]
        if (tid < 256) {
            uint32_t suf     = s_hist[tid];
            uint32_t greater = (tid < 255) ? s_hist[tid + 1] : 0u;
            if (suf >= krem && greater < krem) {
                s_sel[0] = (prefix << 8) | (uint32_t)tid;
                s_sel[1] = krem - greater;     // ties of this digit still needed
            }
        }
        __syncthreads();
    }

    const uint32_t T         = s_sel[0];   // full 32-bit threshold key
    const uint32_t keep_ties = s_sel[1];   // # of elements equal to T to keep

    // ---- gate + write out (16B/lane coalesced stores) ----
    for (int e = 0; e < VEC_PER_THREAD; ++e) {
        uint4 b = reinterpret_cast<uint4*>(s_x)[e * BLOCK + tid];
        uint32_t kk[4] = { f2key(b.x), f2key(b.y), f2key(b.z), f2key(b.w) };
        uint32_t bb[4] = { b.x, b.y, b.z, b.w };
        float o[4];
#pragma unroll
        for (int j = 0; j < 4; ++j) {
            if (kk[j] > T) {
                o[j] = __uint_as_float(bb[j]);
            } else if (kk[j] == T) {
                uint32_t r = atomicAdd(&s_tie, 1u);
                o[j] = (r < keep_ties) ? __uint_as_float(bb[j]) : 0.0f;
            } else {
                o[j] = 0.0f;
            }
        }
        reinterpret_cast<float4*>(outp)[e * BLOCK + tid] =
            make_float4(o[0], o[1], o[2], o[3]);
    }
}

extern "C" void kernel_launch(void* const* d_in, const int* in_sizes, int n_in,
                              void* d_out, int out_size, void* d_ws, size_t ws_size,
                              hipStream_t stream) {
    const float* x    = (const float*)d_in[0];
    const int*   kptr = (const int*)d_in[1];
    float*       out  = (float*)d_out;
    (void)in_sizes; (void)n_in; (void)out_size; (void)d_ws; (void)ws_size;

    topk_gate_kernel<<<N_TOKENS, BLOCK, 0, stream>>>(x, kptr, out);
}